// NLQHead_68135361183835
// MI455X (gfx1250) — compile-verified
//
#include <hip/hip_runtime.h>
#include <math.h>

// ---------------------------------------------------------------------------
// NLQ post-processing head for MI455X (gfx1250, wave32).
//  Phase 1: soft-NMS; scores resident in 256KB LDS of one WGP (320KB avail),
//           filled by one TDM tensor_load_to_lds DMA (TENSORcnt path).
//           Decay skips the ~97% of segments with zero intersection
//           (exp(0)=1 -> bit-exact skip), removing most exp + LDS RMW work.
//  Phase 2: IoU-weighted seg voting via V_WMMA_F32_16X16X4_F32:
//           D = A(16x4 weights) x B(4x16 [1,start,end,0...]) accumulated over N.
//  Phase 3: bitonic sort (desc, stable) of K=512 kept scores.
//  Phase 4: focal + DIoU loss mean reductions (memory-bound, trivial vs 23.3TB/s).
// ---------------------------------------------------------------------------

#define N_SEGS   65536
#define K_KEEP   512
#define SIGMA_INV 2.0f          // 1/0.5
#define MIN_SC   0.001f
#define VOTE_THR 0.75f
#define ALPHA_C  0.25f
#define EPS_C    1e-8f

typedef __attribute__((ext_vector_type(2))) float v2f;
typedef __attribute__((ext_vector_type(8))) float v8f;

// --------------------------- TDM (Tensor Data Mover) -----------------------
#if __has_builtin(__builtin_amdgcn_tensor_load_to_lds)
#define HAVE_TDM 1
typedef __attribute__((ext_vector_type(4))) unsigned int u32x4;
typedef __attribute__((ext_vector_type(4))) int i32x4;
typedef __attribute__((ext_vector_type(8))) int i32x8;

// DMA a rows x cols f32 tile (row stride == cols, fully contiguous) from
// global memory into LDS at byte offset lds_off. D# layout per CDNA5 ISA
// CDNA5 ISA.
__device__ __forceinline__ void tdm_load_f32_2d(const void* gptr,
                                                unsigned lds_off,
                                                unsigned rows, unsigned cols)
{
  unsigned long long ga = (unsigned long long)gptr;
  u32x4 g0;
  g0[0] = 1u;                                          // count=1, user mode
  g0[1] = lds_off;                                     // lds_addr [63:32]
  g0[2] = (unsigned)(ga & 0xffffffffu);                // global_addr lo
  g0[3] = (unsigned)((ga >> 32) & 0x01ffffffu)         // global_addr [56:32]
          | (2u << 30);                                // type=2 ("image")
  i32x8 g1;
  g1[0] = (int)(2u << 16);                             // data_size=2 (4 bytes)
  g1[1] = (int)((cols & 0xffffu) << 16);               // tensor_dim0 [15:0]
  g1[2] = (int)(((cols >> 16) & 0xffffu)               // tensor_dim0 [31:16]
          | ((rows & 0xffffu) << 16));                 // tensor_dim1 [15:0]
  g1[3] = (int)(((rows >> 16) & 0xffffu)               // tensor_dim1 [31:16]
          | ((cols & 0xffffu) << 16));                 // tile_dim0
  g1[4] = (int)(rows & 0xffffu);                       // tile_dim1 (tile_dim2=0)
  g1[5] = (int)cols;                                   // tensor_dim0_stride lo32
  g1[6] = 0;                                           // stride hi / dim1_stride
  g1[7] = 0;
  i32x4 z4 = {0, 0, 0, 0};
#if __clang_major__ >= 23
  i32x8 z8 = {0, 0, 0, 0, 0, 0, 0, 0};
  __builtin_amdgcn_tensor_load_to_lds(g0, g1, z4, z4, z8, 0);
#else
  __builtin_amdgcn_tensor_load_to_lds(g0, g1, z4, z4, 0);
#endif
}
#endif

// ------------------------------ Phase 1 ------------------------------------
__global__ __launch_bounds__(1024) void soft_nms_kernel(
    const float* __restrict__ segs, const float* __restrict__ scores,
    float* __restrict__ kseg, float* __restrict__ kscore)
{
  extern __shared__ float s_sc[];          // N_SEGS floats = 256KB LDS
  __shared__ float red_s[32];
  __shared__ int   red_i[32];
  __shared__ float s_sel[2];
  __shared__ int   s_idx;

  const int tid  = threadIdx.x;
  const int lane = tid & 31;
  const int wid  = tid >> 5;

  // warm L2/WGP$ with segs (emits gfx1250 global_prefetch_b8)
  for (int i = tid * 64; i < 2 * N_SEGS; i += 1024 * 64)
    __builtin_prefetch(&segs[i], 0, 1);

#ifdef HAVE_TDM
  // One TDM DMA moves all 256KB of scores into LDS (2 x 32768 f32 tile since
  // tile_dim0 is 16-bit). Wave 0 issues; TENSORcnt tracks; barrier releases.
  if (wid == 0) {
    tdm_load_f32_2d(scores, (unsigned)(unsigned long long)(uintptr_t)s_sc,
                    2u, (unsigned)(N_SEGS / 2));
    __builtin_amdgcn_s_wait_tensorcnt(0);
  }
#else
  for (int i = tid; i < N_SEGS; i += 1024) s_sc[i] = scores[i];
#endif
  __syncthreads();

  for (int k = 0; k < K_KEEP; ++k) {
    // ---- parallel argmax (tie-break: lowest index, matching jnp.argmax) ----
    float best = -__builtin_inff();
    int   bi   = 0x7fffffff;
    for (int i = tid; i < N_SEGS; i += 1024) {
      float v = s_sc[i];
      if (v > best) { best = v; bi = i; }
    }
    for (int off = 16; off > 0; off >>= 1) {
      float os = __shfl_xor(best, off);
      int   oi = __shfl_xor(bi, off);
      if (os > best || (os == best && oi < bi)) { best = os; bi = oi; }
    }
    if (lane == 0) { red_s[wid] = best; red_i[wid] = bi; }
    __syncthreads();
    if (wid == 0) {
      best = red_s[lane]; bi = red_i[lane];
      for (int off = 16; off > 0; off >>= 1) {
        float os = __shfl_xor(best, off);
        int   oi = __shfl_xor(bi, off);
        if (os > best || (os == best && oi < bi)) { best = os; bi = oi; }
      }
      if (lane == 0) {
        float ns = segs[2 * bi], ne = segs[2 * bi + 1];
        kseg[2 * k] = ns; kseg[2 * k + 1] = ne;
        kscore[k] = best;
        s_sel[0] = ns; s_sel[1] = ne; s_idx = bi;
      }
    }
    __syncthreads();

    // ---- gaussian decay: only overlapping segments change (exp(0)==1) ----
    const float ns = s_sel[0], ne = s_sel[1];
    const int   si = s_idx;
    const float nlen = ne - ns;
    for (int i = tid; i < N_SEGS; i += 1024) {
      float s = segs[2 * i], e = segs[2 * i + 1];
      float inter = fminf(ne, e) - fmaxf(ns, s);
      if (inter > 0.f || i == si) {
        inter = fmaxf(inter, 0.f);
        float uni = nlen + (e - s) - inter;
        float iou = inter / uni;
        float dec = __expf(-(iou * iou) * SIGMA_INV);
        float v   = s_sc[i] * dec;
        s_sc[i] = (i == si) ? -__builtin_inff() : v;
      }
    }
    __syncthreads();
  }
}

// ------------------------------ Phase 2 ------------------------------------
// One block per 16 NMS rows; 8 waves split the N columns. Each wave chains
// V_WMMA_F32_16X16X4_F32 over its column range; cross-wave reduce in LDS.
__global__ __launch_bounds__(256) void seg_vote_wmma_kernel(
    const float* __restrict__ segs, const float* __restrict__ scores,
    const float* __restrict__ kseg, const float* __restrict__ kscore,
    float* __restrict__ refined)
{
  __shared__ float cred[8][32][8];   // per-wave C tiles
  __shared__ float crow[16][3];      // [m][{sum_w, sum_ws, sum_we}]

  const int tid  = threadIdx.x;
  const int lane = tid & 31;
  const int wv   = tid >> 5;
  const int half = lane >> 4;        // lanes 16-31 handle K=2,3
  const int m    = lane & 15;        // row of A / column of B
  const int row  = blockIdx.x * 16 + m;

  const float ns = kseg[2 * row], ne = kseg[2 * row + 1];
  const float nlen = ne - ns;

  v8f c = {0.f, 0.f, 0.f, 0.f, 0.f, 0.f, 0.f, 0.f};

  const int per_wave = N_SEGS / 8;
  const int j_lo = wv * per_wave;
  for (int jb = j_lo; jb < j_lo + per_wave; jb += 4) {
    const int j0 = jb + 2 * half;
    const int j1 = j0 + 1;
    float s0 = segs[2 * j0], e0 = segs[2 * j0 + 1];
    float s1 = segs[2 * j1], e1 = segs[2 * j1 + 1];
    float sc0 = scores[j0],  sc1 = scores[j1];

    // IoU weights for A-matrix (VALU co-executes with matrix pipe)
    float i0 = fmaxf(fminf(ne, e0) - fmaxf(ns, s0), 0.f);
    float iou0 = i0 / (nlen + (e0 - s0) - i0);
    float w0 = (iou0 >= VOTE_THR) ? sc0 * iou0 : 0.f;

    float i1 = fmaxf(fminf(ne, e1) - fmaxf(ns, s1), 0.f);
    float iou1 = i1 / (nlen + (e1 - s1) - i1);
    float w1 = (iou1 >= VOTE_THR) ? sc1 * iou1 : 0.f;

    v2f a = {w0, w1};                 // A[m][k], k = 2*half + {0,1}
    float b0 = (m == 0) ? 1.f : (m == 1) ? s0 : (m == 2) ? e0 : 0.f;
    float b1 = (m == 0) ? 1.f : (m == 1) ? s1 : (m == 2) ? e1 : 0.f;
    v2f b = {b0, b1};                 // B[k][n], n = m

#if __has_builtin(__builtin_amdgcn_wmma_f32_16x16x4_f32)
    c = __builtin_amdgcn_wmma_f32_16x16x4_f32(
        /*neg_a=*/false, a, /*neg_b=*/false, b,
        /*c_mod=*/(short)0, c, /*reuse_a=*/false, /*reuse_b=*/false);
#else
    for (int v = 0; v < 2; ++v) { c[0] += a[v] * b[v]; }
#endif
  }

#pragma unroll
  for (int v = 0; v < 8; ++v) cred[wv][lane][v] = c[v];
  __syncthreads();

  { // reduce 8 wave-tiles; thread tid owns one (lane, vgpr) slot
    const int l = tid >> 3;          // 0..31
    const int v = tid & 7;           // 0..7
    float sum = 0.f;
#pragma unroll
    for (int w = 0; w < 8; ++w) sum += cred[w][l][v];
    const int mm = v + 8 * (l >> 4); // C layout: m = vgpr + 8*(lane>=16)
    const int nn = l & 15;           // n = lane & 15
    if (nn < 3) crow[mm][nn] = sum;
  }
  __syncthreads();

  if (tid < 16) {
    const int r = blockIdx.x * 16 + tid;
    const float sw = crow[tid][0];
    const float rs = crow[tid][1] / sw;
    const float re = crow[tid][2] / sw;
    const bool keep = kscore[r] > MIN_SC;
    refined[2 * r]     = keep ? rs : 0.f;
    refined[2 * r + 1] = keep ? re : 0.f;
  }
}

// ------------------------------ Phase 3 ------------------------------------
__global__ __launch_bounds__(512) void sort_kernel(
    const float* __restrict__ refined, const float* __restrict__ kscore,
    float* __restrict__ out)
{
  __shared__ float ss[512];
  __shared__ int   si[512];
  const int tid = threadIdx.x;
  float sc = kscore[tid];
  sc = (sc > MIN_SC) ? sc : 0.f;
  ss[tid] = sc; si[tid] = tid;
  __syncthreads();

  for (int size = 2; size <= 512; size <<= 1) {
    for (int stride = size >> 1; stride > 0; stride >>= 1) {
      const int p = tid ^ stride;
      if (p > tid) {
        const bool up = ((tid & size) == 0);
        float a = ss[tid], b = ss[p];
        int   ia = si[tid], ib = si[p];
        // "a first" := score desc, tie -> index asc (matches stable argsort(-s))
        const bool a_first = (a > b) || (a == b && ia < ib);
        if (a_first != up) {
          ss[tid] = b; ss[p] = a;
          si[tid] = ib; si[p] = ia;
        }
      }
      __syncthreads();
    }
  }

  const int idx = si[tid];
  out[1024 + tid]    = ss[tid];
  out[2 * tid]       = refined[2 * idx];
  out[2 * tid + 1]   = refined[2 * idx + 1];
}

// ------------------------------ Phase 4 ------------------------------------
__global__ __launch_bounds__(256) void loss_partial_kernel(
    const float* __restrict__ logits, const int* __restrict__ targets,
    const float* __restrict__ op, const float* __restrict__ ot,
    float* __restrict__ fpart, float* __restrict__ dpart)
{
  __shared__ float sf[256], sd[256];
  const int tid = threadIdx.x;
  const int i = blockIdx.x * 256 + tid;

  float x = logits[i];
  float t = (float)targets[i];
  float en  = __expf(-fabsf(x));
  float ls  = fminf(x, 0.f) - log1pf(en);    // log_sigmoid(x)
  float lsn = fminf(-x, 0.f) - log1pf(en);   // log_sigmoid(-x)
  float p  = 1.f / (1.f + __expf(-x));
  float ce = -(t * ls + (1.f - t) * lsn);
  float pt = p * t + (1.f - p) * (1.f - t);
  float om = 1.f - pt;
  float at = ALPHA_C * t + (1.f - ALPHA_C) * (1.f - t);
  float f  = at * ce * om * om;

  float lp = op[2 * i], rp = op[2 * i + 1];
  float lg = ot[2 * i], rg = ot[2 * i + 1];
  float in_ = fminf(rp, rg) + fminf(lp, lg);
  float un  = lp + rp + lg + rg - in_;
  float iou = in_ / fmaxf(un, EPS_C);
  float lc  = fmaxf(lp, lg) + fmaxf(rp, rg);
  float rho = 0.5f * (rp - lp - rg + lg);
  float q   = rho / fmaxf(lc, EPS_C);
  float d   = 1.f - iou + q * q;

  sf[tid] = f; sd[tid] = d;
  __syncthreads();
  for (int s = 128; s > 0; s >>= 1) {
    if (tid < s) { sf[tid] += sf[tid + s]; sd[tid] += sd[tid + s]; }
    __syncthreads();
  }
  if (tid == 0) { fpart[blockIdx.x] = sf[0]; dpart[blockIdx.x] = sd[0]; }
}

__global__ __launch_bounds__(256) void loss_final_kernel(
    const float* __restrict__ fpart, const float* __restrict__ dpart,
    float* __restrict__ out)
{
  __shared__ float sf[256], sd[256];
  const int tid = threadIdx.x;
  sf[tid] = fpart[tid]; sd[tid] = dpart[tid];
  __syncthreads();
  for (int s = 128; s > 0; s >>= 1) {
    if (tid < s) { sf[tid] += sf[tid + s]; sd[tid] += sd[tid + s]; }
    __syncthreads();
  }
  if (tid == 0) {
    out[1536] = sf[0] / (float)N_SEGS;
    out[1537] = sd[0] / (float)N_SEGS;
  }
}

// ------------------------------ launcher -----------------------------------
extern "C" void kernel_launch(void* const* d_in, const int* in_sizes, int n_in,
                              void* d_out, int out_size, void* d_ws, size_t ws_size,
                              hipStream_t stream) {
  const float* segs    = (const float*)d_in[0];
  const float* scores  = (const float*)d_in[1];
  const float* logits  = (const float*)d_in[2];
  const int*   targets = (const int*)d_in[3];
  const float* op      = (const float*)d_in[4];
  const float* ot      = (const float*)d_in[5];
  float* out = (float*)d_out;
  float* ws  = (float*)d_ws;

  float* kseg    = ws;           // 1024 floats
  float* kscore  = ws + 1024;    // 512
  float* refined = ws + 1536;    // 1024
  float* fpart   = ws + 2560;    // 256
  float* dpart   = ws + 2816;    // 256

  const size_t smem_nms = (size_t)N_SEGS * sizeof(float);   // 256 KB
  hipFuncSetAttribute(reinterpret_cast<const void*>(soft_nms_kernel),
                      hipFuncAttributeMaxDynamicSharedMemorySize, (int)smem_nms);

  soft_nms_kernel<<<1, 1024, smem_nms, stream>>>(segs, scores, kseg, kscore);
  seg_vote_wmma_kernel<<<K_KEEP / 16, 256, 0, stream>>>(segs, scores, kseg, kscore, refined);
  sort_kernel<<<1, 512, 0, stream>>>(refined, kscore, out);
  loss_partial_kernel<<<N_SEGS / 256, 256, 0, stream>>>(logits, targets, op, ot, fpart, dpart);
  loss_final_kernel<<<1, 256, 0, stream>>>(fpart, dpart, out);
}